// MoeLayer_58076547776731
// MI455X (gfx1250) — compile-verified
//
#include <hip/hip_runtime.h>
#include <hip/hip_bf16.h>
#include <math.h>

typedef float v2f __attribute__((ext_vector_type(2)));
typedef float v8f __attribute__((ext_vector_type(8)));

#define B_TOK 16384
#define D_DIM 1024
#define E_EXP 8
#define F_DIM 4096
#define C_OUT 7

#define M_TILE 128      // tokens per workgroup tile
#define FB     64       // F (N) chunk per iteration
#define KB     32       // K panel staged in LDS
#define XS_STRIDE 34    // 32 + pad (even -> b64-aligned float2 reads, bank spread)
#define WT_STRIDE 34    // transposed W1 panel: Wt[f][k], same stride properties

// ---------------------------------------------------------------------------
// Kernel 0: zero output + routing counters (graph-capture safe init)
// ---------------------------------------------------------------------------
__global__ void moe_init(float* __restrict__ out, int* __restrict__ cnt) {
    int i = blockIdx.x * blockDim.x + threadIdx.x;
    if (i < E_EXP) cnt[i] = 0;
    if (i < B_TOK * C_OUT) out[i] = 0.0f;
}

// ---------------------------------------------------------------------------
// Kernel 1: gating (one wave per token) + top-2 softmax + routed compaction
// ---------------------------------------------------------------------------
__global__ __launch_bounds__(256)
void moe_gate_route(const float* __restrict__ x,
                    const float* __restrict__ Wg,
                    const float* __restrict__ bg,
                    int* __restrict__ cnt,
                    int* __restrict__ tokList,
                    float* __restrict__ tokW) {
    const int wave  = threadIdx.x >> 5;
    const int lane  = threadIdx.x & 31;
    const int token = blockIdx.x * (blockDim.x >> 5) + wave;
    if (token >= B_TOK) return;

    float acc[E_EXP] = {};
    const float* xr = x + (size_t)token * D_DIM;
    for (int d = lane; d < D_DIM; d += 32) {
        const float xv = xr[d];
        const float* wr = Wg + d * E_EXP;
#pragma unroll
        for (int e = 0; e < E_EXP; ++e) acc[e] += xv * wr[e];
    }
#pragma unroll
    for (int e = 0; e < E_EXP; ++e) {
#pragma unroll
        for (int off = 16; off > 0; off >>= 1)
            acc[e] += __shfl_xor(acc[e], off, 32);
    }
    if (lane == 0) {
        float l0 = -__builtin_inff(), l1 = -__builtin_inff();
        int   i0 = 0, i1 = 0;
#pragma unroll
        for (int e = 0; e < E_EXP; ++e) {
            const float l = acc[e] + bg[e];
            if (l > l0)      { l1 = l0; i1 = i0; l0 = l; i0 = e; }
            else if (l > l1) { l1 = l;  i1 = e; }
        }
        // softmax over the two selected logits (l0 >= l1, numerically stable)
        const float t  = expf(l1 - l0);
        const float w0 = 1.0f / (1.0f + t);
        const float w1 = t * w0;
        int p0 = atomicAdd(&cnt[i0], 1);
        tokList[i0 * B_TOK + p0] = token;
        tokW   [i0 * B_TOK + p0] = w0;
        int p1 = atomicAdd(&cnt[i1], 1);
        tokList[i1 * B_TOK + p1] = token;
        tokW   [i1 * B_TOK + p1] = w1;
    }
}

// ---------------------------------------------------------------------------
// Kernel 2: per-expert fused MLP on routed tokens.
//   h = gelu(Xg @ W1[e] + b1[e]); out[tok] += w * (h @ W2[e] + b2[e])
// WMMA f32 16x16x4; W1 panel transposed in LDS so A and B fragments are both
// single contiguous ds_load_b64s; h stays in registers end-to-end.
// ---------------------------------------------------------------------------
__global__ __launch_bounds__(256)
void moe_expert_mlp(const float* __restrict__ x,
                    const float* __restrict__ W1,
                    const float* __restrict__ b1,
                    const float* __restrict__ W2,
                    const float* __restrict__ b2,
                    const int*   __restrict__ cnt,
                    const int*   __restrict__ tokList,
                    const float* __restrict__ tokW,
                    float* __restrict__ out) {
    const int e   = blockIdx.y;
    const int n_e = cnt[e];
    const int t0  = blockIdx.x * M_TILE;
    if (t0 >= n_e) return;                    // empty tile for this expert

    __shared__ float Xs[M_TILE][XS_STRIDE];   // gathered X panel   [128 x 32]
    __shared__ float Wt[FB][WT_STRIDE];       // W1 panel transposed [64f x 32k]
    __shared__ float W2s[FB][8];              // W2 chunk            [64 x 7]
    __shared__ float outAcc[M_TILE][8];       // per-tile out        [128 x 7]
    __shared__ int   tokS[M_TILE];
    __shared__ float wS[M_TILE];

    const int tid = threadIdx.x;
    if (tid < M_TILE) {
        const int idx   = t0 + tid;
        const int valid = idx < n_e;
        tokS[tid] = valid ? tokList[e * B_TOK + idx] : -1;
        wS[tid]   = valid ? tokW[e * B_TOK + idx]    : 0.0f;
    }
    __syncthreads();

    const int wv   = tid >> 5;
    const int lane = tid & 31;
    const int half = lane >> 4;               // 0: lanes 0-15, 1: lanes 16-31
    const int l15  = lane & 15;
    const int mRow = 16 * wv + l15;           // A-fragment source row

    const float* W1e = W1 + (size_t)e * D_DIM * F_DIM;
    const float* W2e = W2 + (size_t)e * F_DIM * C_OUT;
    const float* b1e = b1 + (size_t)e * F_DIM;

    float outPart[8][C_OUT] = {};             // per-thread h@W2 partials

    for (int f0 = 0; f0 < F_DIM; f0 += FB) {
        v8f acc[4] = {};                      // 16x64 h slab per wave (f32)

        // stage W2 chunk [64 x 7]
        for (int i = tid; i < FB * C_OUT; i += 256) {
            const int fl = i / C_OUT, c = i % C_OUT;
            W2s[fl][c] = W2e[(size_t)(f0 + fl) * C_OUT + c];
        }

        for (int k0 = 0; k0 < D_DIM; k0 += KB) {
            // stage gathered X panel: 128 rows x 32 k, 4 float4 per thread
#pragma unroll
            for (int i = 0; i < 4; ++i) {
                const int linear = tid + 256 * i;       // 0..1023
                const int row = linear >> 3;
                const int c4  = (linear & 7) << 2;
                const int tk  = tokS[row];
                const float4 v = *(const float4*)(x + (size_t)(tk < 0 ? 0 : tk) * D_DIM + k0 + c4);
                Xs[row][c4 + 0] = v.x; Xs[row][c4 + 1] = v.y;
                Xs[row][c4 + 2] = v.z; Xs[row][c4 + 3] = v.w;
            }
            // stage W1 panel transposed: global row-major [k][f] -> LDS [f][k]
#pragma unroll
            for (int i = 0; i < 2; ++i) {
                const int linear = tid + 256 * i;       // 0..511
                const int krow = linear >> 4;           // k within panel
                const int c4   = (linear & 15) << 2;    // f within chunk
                const float4 v = *(const float4*)(W1e + (size_t)(k0 + krow) * F_DIM + f0 + c4);
                Wt[c4 + 0][krow] = v.x; Wt[c4 + 1][krow] = v.y;
                Wt[c4 + 2][krow] = v.z; Wt[c4 + 3][krow] = v.w;
            }
            // prefetch next W1 panel into cache (global_prefetch_b8)
            if (k0 + KB < D_DIM)
                __builtin_prefetch(W1e + (size_t)(k0 + KB + (tid >> 3)) * F_DIM
                                       + f0 + ((tid & 7) << 3), 0, 3);
            __syncthreads();

#pragma unroll
            for (int kk = 0; kk < KB; kk += 4) {
                const int ko = kk + 2 * half;           // K pair this half owns
                const v2f a = *(const v2f*)&Xs[mRow][ko];
#pragma unroll
                for (int sub = 0; sub < 4; ++sub) {
                    const v2f b = *(const v2f*)&Wt[16 * sub + l15][ko];
                    acc[sub] = __builtin_amdgcn_wmma_f32_16x16x4_f32(
                        false, a, false, b, (short)0, acc[sub], false, false);
                }
            }
            __syncthreads();
        }

        // chunk epilogue: bias + tanh-GELU + fold into h@W2 partials (registers)
#pragma unroll
        for (int sub = 0; sub < 4; ++sub) {
            const int f_loc = 16 * sub + l15;
            const float b1v = b1e[f0 + f_loc];
#pragma unroll
            for (int r = 0; r < 8; ++r) {
                const float h  = acc[sub][r] + b1v;
                const float h3 = h * h * h;
                const float g  = 0.5f * h *
                    (1.0f + tanhf(0.7978845608028654f * (h + 0.044715f * h3)));
#pragma unroll
                for (int c = 0; c < C_OUT; ++c)
                    outPart[r][c] += g * W2s[f_loc][c];
            }
        }
        __syncthreads();   // protect W2s/Wt before next chunk rewrites them
    }

    // reduce partials across the 16-lane halves; each (wave,r,half) owns one row
#pragma unroll
    for (int r = 0; r < 8; ++r) {
#pragma unroll
        for (int c = 0; c < C_OUT; ++c) {
            float v = outPart[r][c];
            v += __shfl_xor(v, 1, 32);
            v += __shfl_xor(v, 2, 32);
            v += __shfl_xor(v, 4, 32);
            v += __shfl_xor(v, 8, 32);
            if (l15 == 0) outAcc[16 * wv + r + 8 * half][c] = v;
        }
    }
    __syncthreads();

    // scale by routing weight, add b2, scatter-accumulate to output
    if (tid < M_TILE) {
        const int tk = tokS[tid];
        if (tk >= 0) {
            const float w = wS[tid];
#pragma unroll
            for (int c = 0; c < C_OUT; ++c) {
                const float val = (outAcc[tid][c] + b2[(size_t)e * C_OUT + c]) * w;
                atomicAdd(&out[(size_t)tk * C_OUT + c], val);
            }
        }
    }
}

// ---------------------------------------------------------------------------
extern "C" void kernel_launch(void* const* d_in, const int* in_sizes, int n_in,
                              void* d_out, int out_size, void* d_ws, size_t ws_size,
                              hipStream_t stream) {
    const float* x  = (const float*)d_in[0];   // [B, D]
    // d_in[1] = label (int64) — unused by forward output
    const float* Wg = (const float*)d_in[2];   // [D, E]
    const float* bg = (const float*)d_in[3];   // [E]
    const float* W1 = (const float*)d_in[4];   // [E, D, F]
    const float* b1 = (const float*)d_in[5];   // [E, F]
    const float* W2 = (const float*)d_in[6];   // [E, F, C]
    const float* b2 = (const float*)d_in[7];   // [E, C]
    float* out = (float*)d_out;                // [B, C]

    // workspace layout: counters | token lists | token weights (~1 MB)
    int*   cnt     = (int*)d_ws;                        // E (padded to 16)
    int*   tokList = cnt + 16;                          // E * B
    float* tokW    = (float*)(tokList + E_EXP * B_TOK); // E * B
    (void)ws_size; (void)in_sizes; (void)n_in; (void)out_size;

    moe_init<<<(B_TOK * C_OUT + 255) / 256, 256, 0, stream>>>(out, cnt);

    moe_gate_route<<<B_TOK / 8, 256, 0, stream>>>(x, Wg, bg, cnt, tokList, tokW);

    dim3 grid(B_TOK / M_TILE, E_EXP);   // worst-case tiles per expert; early-exit
    moe_expert_mlp<<<grid, 256, 0, stream>>>(x, W1, b1, W2, b2,
                                             cnt, tokList, tokW, out);
}